// QueryModule_13108240187579
// MI455X (gfx1250) — compile-verified
//
#include <hip/hip_runtime.h>

// Residual VQ, depth 4:  dist = ||r||^2 + ||c||^2 - 2 r.cT  via fp32 WMMA.
// Codebook staged into LDS by the Tensor Data Mover (padded layout via TDM pad).
// Outputs: z_q [131072,64] then maps [4,131072,1024], both fp32, concatenated.

typedef float v2f __attribute__((ext_vector_type(2)));
typedef float v8f __attribute__((ext_vector_type(8)));
typedef unsigned int u32x4 __attribute__((ext_vector_type(4)));
typedef unsigned int u32x8 __attribute__((ext_vector_type(8)));

#define B_ROWS 131072
#define NCODES 1024
#define DIMS 64
#define LDS_STRIDE 68   // 64 + 4 pad: each wave32 b64 B-frag load hits all 64 banks once
#define RVQ_DEPTH 4
#define ROWS_PER_WAVE 16
#define WAVES_PER_BLOCK 16
#define THREADS (WAVES_PER_BLOCK * 32)
#define ROWS_PER_BLOCK (ROWS_PER_WAVE * WAVES_PER_BLOCK)  // 256

__launch_bounds__(THREADS)
__global__ void rvq_kernel(const float* __restrict__ z,
                           const float* __restrict__ cb,
                           float* __restrict__ zq,
                           float* __restrict__ maps) {
  // codebook (later scaled to -2x in place) with padded rows: 283 KB of 320 KB WGP LDS
  __shared__ float lds_cb2[NCODES * LDS_STRIDE];
  __shared__ float lds_cn[NCODES];  // true ||c||^2

  const int tid  = threadIdx.x;
  const int wave = tid >> 5;

  // ---- TDM: DMA the whole 1024x64 f32 codebook into LDS with 4-dword row pad ----
  // D# per CDNA5 ISA (2D tensor -> groups 2/3 omitted).
  if (wave == 0) {
    const unsigned long long ga = (unsigned long long)(uintptr_t)cb;
    const unsigned int lds_off = (unsigned int)(uintptr_t)&lds_cb2[0];  // LDS aperture: low 32 bits

    u32x4 g0;
    g0.x = 1u;                                   // count=1 (valid), load, no gather
    g0.y = lds_off;                              // lds_addr [63:32]
    g0.z = (unsigned int)ga;                     // global_addr [95:64]
    g0.w = (unsigned int)((ga >> 32) & 0x1FFFFFFu) | 0x80000000u;  // addr[56:32] | type=2

    u32x8 g1;
    g1.s0 = (2u << 16)      // data_size = 4 bytes
          | (1u << 20)      // pad_enable
          | (5u << 22)      // pad_interval: every 64 dwords
          | (3u << 25);     // pad_amount: 4 dwords  -> LDS row stride 68 floats
    g1.s1 = (unsigned int)(DIMS) << 16;          // tensor_dim0[15:0] in [79:64]... bits 79:48
    g1.s2 = (unsigned int)(NCODES) << 16;        // tensor_dim1[15:0] at bits 95:80
    g1.s3 = (unsigned int)(DIMS) << 16;          // tile_dim0 at bits 127:112
    g1.s4 = (unsigned int)(NCODES);              // tile_dim1 at bits 143:128
    g1.s5 = (unsigned int)(DIMS);                // tensor_dim0_stride low 32 (bits 191:160)
    g1.s6 = 0u;
    g1.s7 = 0u;

    asm volatile("tensor_load_to_lds %0, %1"
                 :: "s"(g0), "s"(g1)
                 : "memory");
    __builtin_amdgcn_s_wait_tensorcnt(0);
  }
  __syncthreads();

  // ---- norms pass doubles as the (-2)x in-place scale (row-exclusive per thread) ----
  for (int row = tid; row < NCODES; row += THREADS) {
    float* l = lds_cb2 + row * LDS_STRIDE;
    float s = 0.0f;
    #pragma unroll
    for (int i = 0; i < DIMS / 4; ++i) {
      float4 v = *(float4*)(l + 4 * i);
      s += v.x * v.x + v.y * v.y + v.z * v.z + v.w * v.w;
      v.x *= -2.0f; v.y *= -2.0f; v.z *= -2.0f; v.w *= -2.0f;
      *(float4*)(l + 4 * i) = v;
    }
    lds_cn[row] = s;  // computed pre-scale: true ||c||^2
  }
  __syncthreads();

  const int lane = tid & 31;
  const int m    = lane & 15;   // tile row (A) / tile col (B,C) owned by this lane
  const int hi   = lane >> 4;   // half-wave select
  const int koff = hi * 2;      // K offset inside a 16x4 fragment
  const int rowbase = blockIdx.x * ROWS_PER_BLOCK + wave * ROWS_PER_WAVE;
  const int myrow   = rowbase + m;

  // ---- residual = z, held as sixteen 16x4 fp32 A-fragments ----
  // A layout (ISA 7.12.2): lanes 0-15 hold K=4j,4j+1; lanes 16-31 hold K=4j+2,4j+3
  v2f rfrag[16];
  #pragma unroll
  for (int j = 0; j < 16; ++j) {
    float2 v = *(const float2*)(z + (size_t)myrow * DIMS + 4 * j + koff);
    rfrag[j].x = v.x; rfrag[j].y = v.y;
  }

  for (int d = 0; d < RVQ_DEPTH; ++d) {
    // ---- ||residual||^2 per row (halves hold disjoint K, combine via xor-16) ----
    float s = 0.0f;
    #pragma unroll
    for (int j = 0; j < 16; ++j)
      s += rfrag[j].x * rfrag[j].x + rfrag[j].y * rfrag[j].y;
    s += __shfl_xor(s, 16, 32);
    // rn[r] = ||residual||^2 of row (r + 8*hi)  (C/D row layout)
    float rn[8];
    #pragma unroll
    for (int r = 0; r < 8; ++r) rn[r] = __shfl(s, r + 8 * hi, 32);

    // Per-lane running argmin: lane sees columns n == m (mod 16), indices strictly
    // increasing across tiles -> strict '<' inside the loop, no shuffles needed.
    float minv[8];
    int   mini[8];
    #pragma unroll
    for (int r = 0; r < 8; ++r) { minv[r] = 3.4e38f; mini[r] = 0; }

    float* mapd = maps + (size_t)d * B_ROWS * NCODES;
    // running pointers (per-r offset becomes an immediate in the store)
    float*       mp   = mapd + (size_t)(rowbase + 8 * hi) * NCODES + m;
    const float* bptr = lds_cb2 + m * LDS_STRIDE + koff;
    const float* cnp  = lds_cn + m;

    for (int t = 0; t < NCODES / 16; ++t) {
      const int n0 = t * 16;

      // cn issued first: C-init only needs dscnt<=8 (DS returns in order)
      const float cn = *cnp;

      // B fragments (4x16 of -2*codebook^T) from padded LDS
      v2f bfrag[16];
      #pragma unroll
      for (int j = 0; j < 16; ++j) {
        float2 bv = *(const float2*)(bptr + 4 * j);
        bfrag[j].x = bv.x; bfrag[j].y = bv.y;
      }

      // C init: ||r||^2 + ||c||^2  (WMMA adds -2 r.cT on top)
      v8f acc;
      #pragma unroll
      for (int r = 0; r < 8; ++r) acc[r] = rn[r] + cn;

      // 16 chained fp32 WMMAs: acc = A_j * B_j + acc over K=64
      #pragma unroll
      for (int j = 0; j < 16; ++j) {
        acc = __builtin_amdgcn_wmma_f32_16x16x4_f32(
            false, rfrag[j], false, bfrag[j], (short)0, acc, false, false);
      }

      // ---- stream map tile out (non-temporal) + branchless running argmin ----
      #pragma unroll
      for (int r = 0; r < 8; ++r) {
        __builtin_nontemporal_store(acc[r], mp + r * NCODES);
        const float v = acc[r];
        const bool take = v < minv[r];
        mini[r] = take ? (n0 + m) : mini[r];
        minv[r] = take ? v : minv[r];
      }

      mp   += 16;
      bptr += 16 * LDS_STRIDE;
      cnp  += 16;
    }

    // ---- one cross-lane butterfly per depth (branchless, lowest-index ties) ----
    #pragma unroll
    for (int r = 0; r < 8; ++r) {
      float v  = minv[r];
      int  idx = mini[r];
      #pragma unroll
      for (int off = 8; off >= 1; off >>= 1) {
        const float ov = __shfl_xor(v, off, 32);
        const int   oi = __shfl_xor(idx, off, 32);
        const bool take = (ov < v) | ((ov == v) & (oi < idx));
        v   = take ? ov : v;
        idx = take ? oi : idx;
      }
      mini[r] = idx;
    }

    // ---- broadcast argmin to the lane owning each A row ----
    int pred = 0;
    #pragma unroll
    for (int r = 0; r < 8; ++r) {
      const int pLow  = __shfl(mini[r], 0, 32);   // row r      (low half)
      const int pHigh = __shfl(mini[r], 16, 32);  // row r + 8  (high half)
      pred = (m == r)     ? pLow  : pred;
      pred = (m == r + 8) ? pHigh : pred;
    }

    // ---- residual -= codebook[pred]  ==  residual += 0.5 * (-2*codebook)[pred] ----
    const float* csrc = lds_cb2 + (size_t)pred * LDS_STRIDE + koff;
    #pragma unroll
    for (int j = 0; j < 16; ++j) {
      float2 cv = *(const float2*)(csrc + 4 * j);
      rfrag[j].x += 0.5f * cv.x;
      rfrag[j].y += 0.5f * cv.y;
    }
  }

  // ---- z_q = z - final residual ----
  #pragma unroll
  for (int j = 0; j < 16; ++j) {
    float2 zv = *(const float2*)(z + (size_t)myrow * DIMS + 4 * j + koff);
    float2 o;
    o.x = zv.x - rfrag[j].x;
    o.y = zv.y - rfrag[j].y;
    *(float2*)(zq + (size_t)myrow * DIMS + 4 * j + koff) = o;
  }
}

extern "C" void kernel_launch(void* const* d_in, const int* in_sizes, int n_in,
                              void* d_out, int out_size, void* d_ws, size_t ws_size,
                              hipStream_t stream) {
  const float* z  = (const float*)d_in[0];   // [131072, 64] fp32
  const float* cb = (const float*)d_in[1];   // [1024, 64]   fp32
  float* zq   = (float*)d_out;                           // [131072, 64]
  float* maps = (float*)d_out + (size_t)B_ROWS * DIMS;   // [4, 131072, 1024]

  dim3 grid(B_ROWS / ROWS_PER_BLOCK);  // 512 blocks, exact cover
  dim3 block(THREADS);                 // 16 wave32s, 16 rows each
  rvq_kernel<<<grid, block, 0, stream>>>(z, cb, zq, maps);
}